// FAPELoss_481036337738
// MI455X (gfx1250) — compile-verified
//
#include <hip/hip_runtime.h>
#include <math.h>

typedef __attribute__((ext_vector_type(2))) float v2f;
typedef __attribute__((ext_vector_type(8))) float v8f;

#define FAPE_L   1024
#define FAPE_NT  64          // 1024 / 16 tiles per axis
#define CLAMP_D  10.0f
#define INV_UNIT 0.1f
#define FAPE_EPS 1e-8f

// ---------------------------------------------------------------------------
// ws[0] = numerator accumulator, ws[1] = denominator accumulator
// ---------------------------------------------------------------------------
__global__ void fape_init(float* ws) {
    if (threadIdx.x < 2) ws[threadIdx.x] = 0.0f;
}

// Denominator: mask = fm[b,f] * pm[b,p]  =>  sum = sum_b (sum_f fm)(sum_p pm)
__global__ __launch_bounds__(256) void fape_den(const float* __restrict__ fm,
                                                const float* __restrict__ pm,
                                                float* __restrict__ ws) {
    const int b = blockIdx.x;
    const int tid = threadIdx.x;
    float sf = 0.0f, sp = 0.0f;
    for (int i = tid; i < FAPE_L; i += 256) {
        sf += fm[(size_t)b * FAPE_L + i];
        sp += pm[(size_t)b * FAPE_L + i];
    }
    __shared__ float rf[256];
    __shared__ float rp[256];
    rf[tid] = sf; rp[tid] = sp;
    __syncthreads();
    for (int s = 128; s > 0; s >>= 1) {
        if (tid < s) { rf[tid] += rf[tid + s]; rp[tid] += rp[tid + s]; }
        __syncthreads();
    }
    if (tid == 0) atomicAdd(&ws[1], rf[0] * rp[0]);
}

// ---------------------------------------------------------------------------
// Main FAPE kernel. Grid: B * 64 blocks (one per batch x frame-tile),
// 256 threads = 8 wave32s; each wave strides over the 64 position tiles.
//
// diff_i[f,p] = sum_j Rp[f,j,i]*pp[p,j] - sum_j Rt[f,j,i]*tp[p,j] - c[f,i]
// is a K=7 contraction (frames x positions) -> two chained
// V_WMMA_F32_16X16X4_F32 per coordinate i, 6 WMMAs per 16x16 tile.
// Coordinates are processed one at a time through a SINGLE live WMMA
// accumulator, folded immediately into a running |d|^2 (pk_fma), so the
// register allocator never has to shuffle saved accumulators around.
//
// A 16x4 f32 operand layout: lane l holds M = l&15; VGPR0={K0 | K2},
// VGPR1={K1 | K3} selected by l>=16.  B 4x16 symmetric with N along lanes.
// C/D 16x16 f32: VGPR r, lane l -> M = r + 8*(l>=16), N = l&15.
// ---------------------------------------------------------------------------
__global__ __launch_bounds__(256) void fape_main(const float* __restrict__ pfr,
                                                 const float* __restrict__ tfr,
                                                 const float* __restrict__ ppo,
                                                 const float* __restrict__ tpo,
                                                 const float* __restrict__ fm,
                                                 const float* __restrict__ pm,
                                                 float* __restrict__ ws) {
    const int b    = blockIdx.x >> 6;
    const int ftil = blockIdx.x & 63;
    const int tid  = threadIdx.x;
    const int lane = tid & 31;
    const int wave = tid >> 5;          // 0..7
    const int hi   = lane >> 4;         // 0 or 1
    const int lo   = lane & 15;

    // ---- Per-lane frame data (frame index = ftil*16 + (lane&15)) ----
    const int f = ftil * 16 + lo;
    const size_t fbase = ((size_t)b * FAPE_L + f) * 16;

    float Rp[3][3], Rt[3][3], tpv[3], ttv[3], cv[3];
#pragma unroll
    for (int j = 0; j < 3; ++j) {
#pragma unroll
        for (int i = 0; i < 3; ++i) {
            Rp[j][i] = pfr[fbase + j * 4 + i];
            Rt[j][i] = tfr[fbase + j * 4 + i];
        }
        tpv[j] = pfr[fbase + j * 4 + 3];
        ttv[j] = tfr[fbase + j * 4 + 3];
    }
#pragma unroll
    for (int i = 0; i < 3; ++i) {
        cv[i] = (Rp[0][i] * tpv[0] + Rp[1][i] * tpv[1] + Rp[2][i] * tpv[2])
              - (Rt[0][i] * ttv[0] + Rt[1][i] * ttv[1] + Rt[2][i] * ttv[2]);
    }

    // A fragments per output coordinate i:
    //   K0..K2 =  Rp[0..2][i],  K3..K5 = -Rt[0..2][i],  K6 = -c[i], K7 = 0
    v2f A1[3], A2[3];
#pragma unroll
    for (int i = 0; i < 3; ++i) {
        A1[i][0] = hi ?  Rp[2][i] :  Rp[0][i];   // K2 | K0
        A1[i][1] = hi ? -Rt[0][i] :  Rp[1][i];   // K3 | K1
        A2[i][0] = hi ? -cv[i]    : -Rt[1][i];   // K6 | K4
        A2[i][1] = hi ?  0.0f     : -Rt[2][i];   // K7 | K5
    }

    // Per-row accumulators: frames_mask weighting and the /10 scaling are
    // applied once, after the tile loop.
    float acc[8];
#pragma unroll
    for (int r = 0; r < 8; ++r) acc[r] = 0.0f;

    for (int ptil = wave; ptil < FAPE_NT; ptil += 8) {
        const int p = ptil * 16 + lo;
        const size_t pbase = ((size_t)b * FAPE_L + p) * 3;
        const float ppx = ppo[pbase + 0], ppy = ppo[pbase + 1], ppz = ppo[pbase + 2];
        const float tpx = tpo[pbase + 0], tpy = tpo[pbase + 1], tpz = tpo[pbase + 2];
        const float pmv = pm[(size_t)b * FAPE_L + p];

        // B fragments: K0..K2 = pred_pos, K3..K5 = target_pos, K6 = 1, K7 = 0
        v2f B1, B2;
        B1[0] = hi ? ppz  : ppx;     // K2 | K0
        B1[1] = hi ? tpx  : ppy;     // K3 | K1
        B2[0] = hi ? 1.0f : tpy;     // K6 | K4
        B2[1] = hi ? 0.0f : tpz;     // K7 | K5

        const v8f z = {};
        v8f w, d2;

        // ---- coordinate 0: d2 = dx*dx ----
        w = __builtin_amdgcn_wmma_f32_16x16x4_f32(false, A2[0], false, B2, (short)0, z, false, false);
        w = __builtin_amdgcn_wmma_f32_16x16x4_f32(false, A1[0], false, B1, (short)0, w, false, false);
#pragma unroll
        for (int r = 0; r < 8; ++r) d2[r] = w[r] * w[r];

        // ---- coordinate 1: d2 += dy*dy ----
        w = __builtin_amdgcn_wmma_f32_16x16x4_f32(false, A2[1], false, B2, (short)0, z, false, false);
        w = __builtin_amdgcn_wmma_f32_16x16x4_f32(false, A1[1], false, B1, (short)0, w, false, false);
#pragma unroll
        for (int r = 0; r < 8; ++r) d2[r] = fmaf(w[r], w[r], d2[r]);

        // ---- coordinate 2: d2 += dz*dz ----
        w = __builtin_amdgcn_wmma_f32_16x16x4_f32(false, A2[2], false, B2, (short)0, z, false, false);
        w = __builtin_amdgcn_wmma_f32_16x16x4_f32(false, A1[2], false, B1, (short)0, w, false, false);
#pragma unroll
        for (int r = 0; r < 8; ++r) d2[r] = fmaf(w[r], w[r], d2[r]);

        // error = min(sqrt(|d|^2 + eps), 10); weight by positions_mask now,
        // frames_mask and 1/10 later. Raw v_sqrt_f32 (TRANS, co-executes):
        // argument is always a positive normal (>= 1e-8), so the precise-sqrt
        // range scaling + Newton refinement of the libcall is waste.
#pragma unroll
        for (int r = 0; r < 8; ++r) {
            float e = __builtin_amdgcn_sqrtf(d2[r] + FAPE_EPS);
            e = fminf(e, CLAMP_D);
            acc[r] += e * pmv;
        }
    }

    // Apply per-row frames_mask and the 1/LOSS_UNIT_DISTANCE scale once.
    float partial = 0.0f;
#pragma unroll
    for (int r = 0; r < 8; ++r) {
        const float fmr = fm[(size_t)b * FAPE_L + ftil * 16 + r + 8 * hi];
        partial += acc[r] * fmr;
    }
    partial *= INV_UNIT;

    // Block reduction, then one float atomic per block
    __shared__ float red[256];
    red[tid] = partial;
    __syncthreads();
    for (int s = 128; s > 0; s >>= 1) {
        if (tid < s) red[tid] += red[tid + s];
        __syncthreads();
    }
    if (tid == 0) atomicAdd(&ws[0], red[0]);
}

__global__ void fape_fin(const float* __restrict__ ws, float* __restrict__ out) {
    if (threadIdx.x == 0) out[0] = ws[0] / (ws[1] + FAPE_EPS);
}

// ---------------------------------------------------------------------------
extern "C" void kernel_launch(void* const* d_in, const int* in_sizes, int n_in,
                              void* d_out, int out_size, void* d_ws, size_t ws_size,
                              hipStream_t stream) {
    const float* pred_frames   = (const float*)d_in[0];
    const float* target_frames = (const float*)d_in[1];
    const float* pred_pos      = (const float*)d_in[2];
    const float* target_pos    = (const float*)d_in[3];
    const float* frames_mask   = (const float*)d_in[4];
    const float* pos_mask      = (const float*)d_in[5];
    float* out = (float*)d_out;
    float* ws  = (float*)d_ws;

    const int B = 8;

    fape_init<<<1, 32, 0, stream>>>(ws);
    fape_den<<<B, 256, 0, stream>>>(frames_mask, pos_mask, ws);
    fape_main<<<B * FAPE_NT, 256, 0, stream>>>(pred_frames, target_frames,
                                               pred_pos, target_pos,
                                               frames_mask, pos_mask, ws);
    fape_fin<<<1, 32, 0, stream>>>(ws, out);
}